// SFDiVeQ_19774029430967
// MI455X (gfx1250) — compile-verified
//
#include <hip/hip_runtime.h>
#include <math.h>

#define B_ROWS 16384
#define KCODE  512
#define DDIM   256
#define NSEG   511

typedef __attribute__((ext_vector_type(2))) float v2f;
typedef __attribute__((ext_vector_type(8))) float v8f;

// ---------------- Kernel A: per-segment scalars ----------------
__global__ void seg_scalars_kernel(const float* __restrict__ cb,
                                   float* __restrict__ sls_o,
                                   float* __restrict__ inv_o,
                                   float* __restrict__ cis_o,
                                   float* __restrict__ cc_o) {
    int i = blockIdx.x * blockDim.x + threadIdx.x;
    if (i >= NSEG) return;
    const float4* a = (const float4*)(cb + i * DDIM);
    const float4* b = (const float4*)(cb + (i + 1) * DDIM);
    float s_ss = 0.f, s_cs = 0.f, s_cc = 0.f;
#pragma unroll 8
    for (int e = 0; e < DDIM / 4; ++e) {
        float4 ca = a[e], cn = b[e];
        float sx = cn.x - ca.x, sy = cn.y - ca.y, sz = cn.z - ca.z, sw = cn.w - ca.w;
        s_ss += sx * sx + sy * sy + sz * sz + sw * sw;
        s_cs += ca.x * sx + ca.y * sy + ca.z * sz + ca.w * sw;
        s_cc += ca.x * ca.x + ca.y * ca.y + ca.z * ca.z + ca.w * ca.w;
    }
    float v = s_ss + 1e-8f;
    sls_o[i] = v;
    inv_o[i] = 1.0f / v;
    cis_o[i] = s_cs;
    cc_o[i]  = s_cc;
}

// ---------------- Kernel B: per-row |z|^2 ----------------
__global__ void zz_kernel(const float* __restrict__ z, float* __restrict__ zz_o) {
    int r = blockIdx.x * blockDim.x + threadIdx.x;
    if (r >= B_ROWS) return;
    const float4* zr = (const float4*)(z + r * DDIM);
    float s = 0.f;
#pragma unroll 8
    for (int e = 0; e < DDIM / 4; ++e) {
        float4 v = zr[e];
        s += v.x * v.x + v.y * v.y + v.z * v.z + v.w * v.w;
    }
    zz_o[r] = s;
}

// ---------------- Main: fused WMMA GEMM + segment argmin + gather ----------------
// Block = 128 threads (4 waves). Each wave owns TWO 16-row M-tiles (32 rows) and
// sweeps all 32 column tiles of G = z @ codebook^T in groups of 4, so each B
// fragment feeds 2 WMMAs and each A fragment feeds 4 (0.75 loads per WMMA).
// Distance/argmin scan is folded into registers via LDS lane-permutes.
__global__ __launch_bounds__(128) void sfdiveq_main(
    const float* __restrict__ z,
    const float* __restrict__ cb,
    const float* __restrict__ sls_arr,
    const float* __restrict__ inv_arr,
    const float* __restrict__ cis_arr,
    const float* __restrict__ cc_arr,
    const float* __restrict__ zz_arr,
    float* __restrict__ out_bp,
    float* __restrict__ out_idx,
    float* __restrict__ out_t,
    float* __restrict__ out_dist)
{
    const int lane = threadIdx.x & 31;
    const int wave = threadIdx.x >> 5;
    const int half = lane >> 4;   // ISA 16x16 layout: lanes 0-15 / 16-31 halves
    const int ln   = lane & 15;
    const int row0 = blockIdx.x * 128 + wave * 32;   // 32 rows per wave

    // A fragments (16x4 f32): lane holds z[rowbase+ln][4k + 2*half + {0,1}]
    const float* zbase0 = z + (row0 +  0 + ln) * DDIM + 2 * half;
    const float* zbase1 = z + (row0 + 16 + ln) * DDIM + 2 * half;

    float carry[2][8];
    float bestd[2][8]; int besti[2][8]; float bestt[2][8];
#pragma unroll
    for (int m = 0; m < 2; ++m)
#pragma unroll
        for (int j = 0; j < 8; ++j) {
            carry[m][j] = 0.f; bestd[m][j] = 3.4e38f; besti[m][j] = 0; bestt[m][j] = 0.f;
        }

    const v8f vzero = {0.f, 0.f, 0.f, 0.f, 0.f, 0.f, 0.f, 0.f};

    for (int ctg = 0; ctg < 8; ++ctg) {       // 8 groups x 4 col-tiles = 512 cols
        v8f acc[2][4];
#pragma unroll
        for (int c = 0; c < 4; ++c) { acc[0][c] = vzero; acc[1][c] = vzero; }

#pragma unroll 2
        for (int k = 0; k < 64; ++k) {        // K-dim: 256 / 4 per WMMA
            v2f a0 = *(const v2f*)(zbase0 + 4 * k);
            v2f a1 = *(const v2f*)(zbase1 + 4 * k);
#pragma unroll
            for (int c = 0; c < 4; ++c) {
                const int col = (ctg * 4 + c) * 16 + ln;   // B fragment: codebook row = G column
                v2f bf = *(const v2f*)(cb + col * DDIM + 4 * k + 2 * half);
                acc[0][c] = __builtin_amdgcn_wmma_f32_16x16x4_f32(
                    false, a0, false, bf, (short)0, acc[0][c], false, false);
                acc[1][c] = __builtin_amdgcn_wmma_f32_16x16x4_f32(
                    false, a1, false, bf, (short)0, acc[1][c], false, false);
            }
        }

        // In-register distance scan. Lane ln handles segment i = 16*ct + ln - 1:
        //   G_i from lane ln-1 (carry across tiles), G_{i+1} = own value.
#pragma unroll
        for (int c = 0; c < 4; ++c) {
            const int ct = ctg * 4 + c;
            const int i  = ct * 16 + ln - 1;
            const int ic = (i < 0) ? 0 : i;
            const float sls  = sls_arr[ic];
            const float isls = inv_arr[ic];
            const float cis  = cis_arr[ic];
            const float ccv  = cc_arr[ic];
            const bool valid = (i >= 0);
#pragma unroll
            for (int m = 0; m < 2; ++m)
#pragma unroll
                for (int j = 0; j < 8; ++j) {
                    float g1 = acc[m][c][j];              // G[row, 16ct+ln] = G_{i+1}
                    float g0 = __shfl_up(g1, 1, 16);      // G_i from previous lane
                    if (ln == 0) g0 = carry[m][j];
                    float dotv = g1 - g0 - cis;
                    float t = dotv * isls;
                    t = t < 0.f ? 0.f : (t > 1.f ? 1.f : t);
                    // d' = -2*G_i + cc - 2*t*dot + t^2*sls   (zz added at the end)
                    float d = fmaf(-2.f, g0, ccv) + t * fmaf(t, sls, -2.f * dotv);
                    if (valid && d < bestd[m][j]) { bestd[m][j] = d; besti[m][j] = i; bestt[m][j] = t; }
                    carry[m][j] = __shfl(g1, 15, 16);     // lane-15 value for next tile
                }
        }
    }

    // Per-row argmin across the 16 lanes of each half (tie-break: lowest index).
#pragma unroll
    for (int m = 0; m < 2; ++m)
#pragma unroll
        for (int j = 0; j < 8; ++j) {
            float d  = bestd[m][j];
            int   ii = besti[m][j];
            float tt = bestt[m][j];
#pragma unroll
            for (int off = 8; off; off >>= 1) {
                float d2 = __shfl_xor(d,  off, 16);
                int   i2 = __shfl_xor(ii, off, 16);
                float t2 = __shfl_xor(tt, off, 16);
                if (d2 < d || (d2 == d && i2 < ii)) { d = d2; ii = i2; tt = t2; }
            }
            if (ln == j) {
                const int r = row0 + m * 16 + j + half * 8;  // C-layout: M = vgpr(+8 hi half)
                const float dist = sqrtf(fmaxf(d + zz_arr[r], 0.f));
                out_idx[r]  = (float)ii;
                out_t[r]    = tt;
                out_dist[r] = dist;
                const float4* ca = (const float4*)(cb + ii * DDIM);
                const float4* cn = (const float4*)(cb + (ii + 1) * DDIM);
                float4* op = (float4*)(out_bp + (size_t)r * DDIM);
#pragma unroll 4
                for (int e = 0; e < DDIM / 4; ++e) {
                    float4 x = ca[e], y = cn[e], o;
                    o.x = fmaf(tt, y.x - x.x, x.x);
                    o.y = fmaf(tt, y.y - x.y, x.y);
                    o.z = fmaf(tt, y.z - x.z, x.z);
                    o.w = fmaf(tt, y.w - x.w, x.w);
                    op[e] = o;
                }
            }
        }
}

extern "C" void kernel_launch(void* const* d_in, const int* in_sizes, int n_in,
                              void* d_out, int out_size, void* d_ws, size_t ws_size,
                              hipStream_t stream) {
    const float* z  = (const float*)d_in[0];   // (B, D) f32
    const float* cb = (const float*)d_in[1];   // (K, D) f32

    float* ws  = (float*)d_ws;                 // 512-stride slots for alignment
    float* sls = ws + 0;
    float* inv = ws + 512;
    float* cis = ws + 1024;
    float* ccv = ws + 1536;
    float* zz  = ws + 2048;                    // B floats

    float* out      = (float*)d_out;
    float* out_bp   = out;                     // (B, D)
    float* out_idx  = out + (size_t)B_ROWS * DDIM;
    float* out_t    = out_idx + B_ROWS;
    float* out_dist = out_t + B_ROWS;

    seg_scalars_kernel<<<(NSEG + 255) / 256, 256, 0, stream>>>(cb, sls, inv, cis, ccv);
    zz_kernel<<<B_ROWS / 256, 256, 0, stream>>>(z, zz);
    sfdiveq_main<<<B_ROWS / 128, 128, 0, stream>>>(
        z, cb, sls, inv, cis, ccv, zz, out_bp, out_idx, out_t, out_dist);
}